// QuantumKernelAttentionHadamard_57690000720069
// MI455X (gfx1250) — compile-verified
//
#include <hip/hip_runtime.h>
#include <hip/hip_bf16.h>
#include <math.h>

// Problem constants (from reference): B,L,S,H,D,P = 2,1024,1024,8,64,4
constexpr int Bc = 2;
constexpr int Lc = 1024;
constexpr int Sc = 1024;
constexpr int Hc = 8;
constexpr int Dc = 64;
constexpr int HD = Hc * Dc;     // row stride (floats) between consecutive s in values
constexpr int NF = 16;          // 2^P feature dimension
constexpr int PST = 20;         // P-tile LDS row stride (pad)
constexpr int VST = 72;         // V-tile LDS row stride (pad: half-wave bank split)
constexpr int KST = 18;         // fk-tile LDS row stride (pad, keeps 8B align)
constexpr int NT = Sc / 16;     // 64 S-tiles

typedef __attribute__((ext_vector_type(2))) float v2f;
typedef __attribute__((ext_vector_type(8))) float v8f;
typedef __attribute__((ext_vector_type(4))) int v4i;

static __device__ __forceinline__ v8f wmma_f32_16x16x4(v2f a, v2f b, v8f c) {
  return __builtin_amdgcn_wmma_f32_16x16x4_f32(
      false, a, false, b, (short)0, c, false, false);
}

// ---- CDNA5 async global->LDS copies (ASYNCcnt-tracked) ---------------------
static __device__ __forceinline__ void async_copy_b32(const float* g, float* l) {
#if __has_builtin(__builtin_amdgcn_global_load_async_to_lds_b32)
  __builtin_amdgcn_global_load_async_to_lds_b32(
      (__attribute__((address_space(1))) int*)g,
      (__attribute__((address_space(3))) int*)l, 0, 0);
#else
  unsigned laddr = (unsigned)(unsigned long long)l;  // low 32 bits = LDS offset
  asm volatile("global_load_async_to_lds_b32 %0, %1, off" ::"v"(laddr), "v"(g)
               : "memory");
#endif
}
static __device__ __forceinline__ void async_copy_b128(const float* g, float* l) {
#if __has_builtin(__builtin_amdgcn_global_load_async_to_lds_b128)
  __builtin_amdgcn_global_load_async_to_lds_b128(
      (__attribute__((address_space(1))) v4i*)g,
      (__attribute__((address_space(3))) v4i*)l, 0, 0);
#else
  unsigned laddr = (unsigned)(unsigned long long)l;
  asm volatile("global_load_async_to_lds_b128 %0, %1, off" ::"v"(laddr), "v"(g)
               : "memory");
#endif
}
static __device__ __forceinline__ void wait_async0() {
#if __has_builtin(__builtin_amdgcn_s_wait_asynccnt)
  __builtin_amdgcn_s_wait_asynccnt(0);
#else
  asm volatile("s_wait_asynccnt 0x0" ::: "memory");
#endif
}

// ---------------------------------------------------------------------------
// Kernel 1: feature map.  x:[B,N,H,D] row r = (b*N+n)*H+h  ->  feat:[B,H,N,16]
// a_p = tanh(x.W[:,p]+bias_p)*pi/2 ; feat[t] = prod_p (bit_p(t)? sin:cos)(a_p)
// ---------------------------------------------------------------------------
__global__ __launch_bounds__(256)
void qka_feat_kernel(const float* __restrict__ x, const float* __restrict__ W,
                     const float* __restrict__ bias, float* __restrict__ feat,
                     int nrows, float scale) {
  int r = blockIdx.x * blockDim.x + threadIdx.x;
  if (r >= nrows) return;
  const int h = r % Hc;
  const int n = (r / Hc) % Lc;
  const int b = r / (Hc * Lc);

  const float* xr = x + (size_t)r * Dc;
  float4 acc = make_float4(bias[0], bias[1], bias[2], bias[3]);
#pragma unroll 8
  for (int d = 0; d < Dc; ++d) {
    float xv = xr[d];
    float4 w = ((const float4*)W)[d];  // W is [D,4] row-major
    acc.x += xv * w.x; acc.y += xv * w.y; acc.z += xv * w.z; acc.w += xv * w.w;
  }
  const float HPI = 1.57079632679489662f;  // pi/2 (half-angle folded in)
  float a0 = tanhf(acc.x) * HPI, a1 = tanhf(acc.y) * HPI;
  float a2 = tanhf(acc.z) * HPI, a3 = tanhf(acc.w) * HPI;
  float c0 = __cosf(a0), s0 = __sinf(a0);
  float c1 = __cosf(a1), s1 = __sinf(a1);
  float c2 = __cosf(a2), s2 = __sinf(a2);
  float c3 = __cosf(a3), s3 = __sinf(a3);

  float f01[4] = {c0 * c1, s0 * c1, c0 * s1, s0 * s1};
  float f2[8];
#pragma unroll
  for (int t = 0; t < 4; ++t) { f2[t] = f01[t] * c2; f2[t + 4] = f01[t] * s2; }
  float f[16];
#pragma unroll
  for (int t = 0; t < 8; ++t) { f[t] = f2[t] * c3 * scale; f[t + 8] = f2[t] * s3 * scale; }

  float* fo = feat + ((size_t)(b * Hc + h) * Lc + n) * NF;
#pragma unroll
  for (int t = 0; t < 4; ++t)
    ((float4*)fo)[t] = make_float4(f[4 * t], f[4 * t + 1], f[4 * t + 2], f[4 * t + 3]);
}

// ---------------------------------------------------------------------------
// Kernel 2: flash attention, WMMA f32 16x16x4.
//  - one wave per 16-row L tile; all 8 waves of a block share (b,h)
//  - block double-buffers fk & V tiles in LDS via async global->LDS copies
//  - scores bounded by 1/8 => softmax without running max; per-lane partial
//    denominators reduced once after the loop
// ---------------------------------------------------------------------------
__global__ __launch_bounds__(256)
void qka_attn_kernel(const float* __restrict__ fq,   // [B,H,L,16] (pre-scaled 1/8)
                     const float* __restrict__ fk,   // [B,H,S,16]
                     const float* __restrict__ V,    // [B,S,H,D]
                     float* __restrict__ ctx_out,    // [B,L,H,D]
                     float* __restrict__ attn_out) { // [B,H,L,S]
  __shared__ float vbuf[2][16 * VST];
  __shared__ float kbuf[2][16 * KST];
  __shared__ float pbuf[8][16 * PST];

  const int tid    = threadIdx.x;
  const int lane   = tid & 31;
  const int wave   = tid >> 5;
  const int wid    = blockIdx.x * 8 + wave;
  const int lt     = wid % (Lc / 16);
  const int bh     = wid / (Lc / 16);   // b*H + h (uniform across the block)
  const int h      = bh % Hc;
  const int b      = bh / Hc;
  const int l0     = lt * 16;
  const int lane16 = lane & 15;
  const int hi     = lane >> 4;

  const float* fqb = fq + ((size_t)bh * Lc + l0) * NF;
  const float* fkb = fk + (size_t)bh * Sc * NF;
  const float* Vb  = V + (size_t)b * Sc * HD + (size_t)h * Dc;

  // A operand (fq tile 16x16) in native WMMA A layout (4 chunks of 16x4).
  v2f afq[4];
  {
    const float* ap = fqb + lane16 * NF + hi * 2;
#pragma unroll
    for (int c = 0; c < 4; ++c) afq[c] = *(const v2f*)(ap + 4 * c);
  }

  const v8f vzero = {0.f, 0.f, 0.f, 0.f, 0.f, 0.f, 0.f, 0.f};
  v8f ctx[4] = {vzero, vzero, vzero, vzero};
  float lsum[8];
#pragma unroll
  for (int i = 0; i < 8; ++i) lsum[i] = 0.f;

  float* pw = pbuf[wave];

  // cooperative prefetch of tile st into buffer buf (one b128 + one b32/thread)
  const int prow = tid >> 4, pseg = tid & 15;
  auto issue_prefetch = [&](int st, int buf) {
    const float* vg = Vb + (size_t)(st * 16 + prow) * HD + pseg * 4;
    async_copy_b128(vg, &vbuf[buf][prow * VST + pseg * 4]);
    const float* kg = fkb + (size_t)st * 16 * NF + tid;  // tile is contiguous
    async_copy_b32(kg, &kbuf[buf][prow * KST + pseg]);
  };

  issue_prefetch(0, 0);

  for (int st = 0; st < NT; ++st) {
    const int cur = st & 1, nxt = cur ^ 1;
    wait_async0();
    __syncthreads();  // prefetched tiles visible; previous buffer free
    if (st + 1 < NT) issue_prefetch(st + 1, nxt);

    // scores = fq . fk  (B operand straight out of padded LDS)
    v2f bfk[4];
    {
      const float* kp = &kbuf[cur][lane16 * KST + hi * 2];
#pragma unroll
      for (int c = 0; c < 4; ++c) bfk[c] = *(const v2f*)(kp + 4 * c);
    }
    v8f sc = vzero;
#pragma unroll
    for (int c = 0; c < 4; ++c) sc = wmma_f32_16x16x4(afq[c], bfk[c], sc);

    // |score| <= 1/8: softmax without max; per-lane partial denominators.
    float p[8];
#pragma unroll
    for (int i = 0; i < 8; ++i) { p[i] = __expf(sc[i]); lsum[i] += p[i]; }

    // Re-layout P (C/D layout -> A layout) through wave-private LDS.
    asm volatile("" ::: "memory");
#pragma unroll
    for (int i = 0; i < 8; ++i) pw[(i + 8 * hi) * PST + lane16] = p[i];
    asm volatile("s_wait_dscnt 0x0" ::: "memory");
    v2f pa[4];
    {
      const float* lp = &pw[lane16 * PST + hi * 2];
#pragma unroll
      for (int c = 0; c < 4; ++c) pa[c] = *(const v2f*)(lp + 4 * c);
    }
    asm volatile("s_wait_dscnt 0x0" ::: "memory");

    // ctx += P(16x16) . V(16x64), V B-operand from padded LDS tile.
    const float* vb0 = &vbuf[cur][0];
#pragma unroll
    for (int c = 0; c < 4; ++c) {
      const float* vrow = vb0 + (4 * c + 2 * hi) * VST;
#pragma unroll
      for (int t = 0; t < 4; ++t) {
        v2f vv;
        vv.x = vrow[t * 16 + lane16];
        vv.y = vrow[VST + t * 16 + lane16];
        ctx[t] = wmma_f32_16x16x4(pa[c], vv, ctx[t]);
      }
    }
  }

  // One-time 16-lane reduction of the softmax denominators.
  float invl[8];
#pragma unroll
  for (int i = 0; i < 8; ++i) {
    float rs = lsum[i];
    rs += __shfl_xor(rs, 1, 32);
    rs += __shfl_xor(rs, 2, 32);
    rs += __shfl_xor(rs, 4, 32);
    rs += __shfl_xor(rs, 8, 32);
    invl[i] = 1.f / rs;
  }
#pragma unroll
  for (int t = 0; t < 4; ++t)
#pragma unroll
    for (int i = 0; i < 8; ++i) ctx[t][i] *= invl[i];

  // Store context: [B,L,H,D]
  float* cb = ctx_out + ((size_t)(b * Lc + l0 + 8 * hi) * Hc + h) * Dc + lane16;
#pragma unroll
  for (int i = 0; i < 8; ++i)
#pragma unroll
    for (int t = 0; t < 4; ++t) cb[(size_t)i * HD + t * 16] = ctx[t][i];

  // Pass 2: recompute scores (4 wmma/tile), emit normalized weights [B,H,L,S].
  float* ab = attn_out + ((size_t)bh * Lc + l0 + 8 * hi) * Sc + lane16;
  for (int st = 0; st < NT; ++st) {
    const int s0 = st * 16;
    v2f bfk[4];
    {
      const float* kp = fkb + (size_t)(s0 + lane16) * NF + hi * 2;
#pragma unroll
      for (int c = 0; c < 4; ++c) bfk[c] = *(const v2f*)(kp + 4 * c);
    }
    v8f sc = vzero;
#pragma unroll
    for (int c = 0; c < 4; ++c) sc = wmma_f32_16x16x4(afq[c], bfk[c], sc);
#pragma unroll
    for (int i = 0; i < 8; ++i)
      ab[(size_t)i * Sc + s0] = __expf(sc[i]) * invl[i];
  }
}

// ---------------------------------------------------------------------------
extern "C" void kernel_launch(void* const* d_in, const int* in_sizes, int n_in,
                              void* d_out, int out_size, void* d_ws, size_t ws_size,
                              hipStream_t stream) {
  const float* q  = (const float*)d_in[0];
  const float* k  = (const float*)d_in[1];
  const float* v  = (const float*)d_in[2];
  const float* Wq = (const float*)d_in[3];
  const float* bq = (const float*)d_in[4];
  const float* Wk = (const float*)d_in[5];
  const float* bk = (const float*)d_in[6];

  float* out      = (float*)d_out;
  float* ctx_out  = out;                              // [B,L,H,D]
  float* attn_out = out + (size_t)Bc * Lc * Hc * Dc;  // [B,H,L,S]

  float* fq = (float*)d_ws;                           // [B,H,L,16]
  float* fk = fq + (size_t)Bc * Hc * Lc * NF;         // [B,H,S,16]

  const int nrows = Bc * Lc * Hc;  // 16384 (L == S)
  qka_feat_kernel<<<(nrows + 255) / 256, 256, 0, stream>>>(q, Wq, bq, fq, nrows, 0.125f);
  qka_feat_kernel<<<(nrows + 255) / 256, 256, 0, stream>>>(k, Wk, bk, fk, nrows, 1.0f);

  const int waves = Bc * Hc * (Lc / 16);  // 1024 waves, 8 per block
  qka_attn_kernel<<<waves / 8, 256, 0, stream>>>(fq, fk, v, ctx_out, attn_out);
}